// Morphology_21122649162174
// MI455X (gfx1250) — compile-verified
//
#include <hip/hip_runtime.h>
#include <math.h>

// Morphological dilation2d (max-plus conv), f32.
// x: (4,8,512,512), weight: (8,8,5,5), out: (4,8,512,512)
// out[n,o,h,w] = max_{c,dy,dx} ( x_zeropad[n,c,h+dy,w+dx] + w[o,c,dy,dx] )

#define KK      5
#define PADR    2
#define CIN     8
#define COUT    8
#define HH      512
#define WW      512
#define TILE_W  64
#define TILE_H  16
#define XROWS   (TILE_H + KK - 1)        // 20
#define XCOLS   (TILE_W + KK - 1)        // 68 (272B rows -> 16B aligned)
#define XSL     XCOLS                    // LDS row stride in floats
#define NXTOT   (CIN * XROWS * XSL)      // 10880 floats = 42.5 KB
#define NWTOT   (COUT * CIN * KK * KK)   // 1600 floats  = 6.25 KB

// gfx1250 async global->LDS DMA (ASYNCcnt-tracked). Builtin signature on this
// toolchain (probed via diagnostics): (int AS1*, int AS3*, imm off, imm cpol).
#if defined(__gfx1250__) && __has_builtin(__builtin_amdgcn_global_load_async_to_lds_b32)
#define USE_ASYNC_LDS 1
#define AS1CAST(p) ((__attribute__((address_space(1))) int*)(p))
#define AS3CAST(p) ((__attribute__((address_space(3))) int*)(p))
#else
#define USE_ASYNC_LDS 0
#endif

__global__ __launch_bounds__(256)
void dilation2d_maxplus(const float* __restrict__ x,
                        const float* __restrict__ wgt,
                        float* __restrict__ out) {
    __shared__ float sx[NXTOT];
    __shared__ float sw[NWTOT];

    const int tid  = threadIdx.x;
    const int n    = blockIdx.z;
    const int row0 = blockIdx.y * TILE_H;
    const int col0 = blockIdx.x * TILE_W;

    // gfx1250 prefetch path (global_prefetch_b8): warm weights in L2.
    if (tid < 50) __builtin_prefetch(wgt + tid * 32, 0, 0);

    // ---- stage x tile into LDS (async DMA) with zero-filled halo ----
    const float* xn = x + (size_t)n * CIN * HH * WW;
    for (int i = tid; i < NXTOT; i += 256) {
        int c   = i / (XROWS * XSL);
        int rem = i - c * (XROWS * XSL);
        int r   = rem / XSL;
        int col = rem - r * XSL;
        int gr  = row0 - PADR + r;
        int gc  = col0 - PADR + col;
        size_t goff = (size_t)c * (HH * WW) + (size_t)gr * WW + gc;
        bool inr = ((unsigned)gr < HH) && ((unsigned)gc < WW);
#if USE_ASYNC_LDS
        if (inr) {
            __builtin_amdgcn_global_load_async_to_lds_b32(
                AS1CAST(const_cast<float*>(xn) + goff), AS3CAST(sx + i), 0, 0);
        } else {
            sx[i] = 0.0f;                 // zero padding (semantics!)
        }
#else
        float v = 0.0f;
        if (inr) v = xn[goff];
        sx[i] = v;
#endif
    }
    // ---- stage weights, relayout [o][c][dy][dx] -> [c][dy][dx][o] ----
    for (int i = tid; i < NWTOT; i += 256) {
        int o = i & 7;
        int t = i >> 3;                   // c*25 + dy*5 + dx
#if USE_ASYNC_LDS
        __builtin_amdgcn_global_load_async_to_lds_b32(
            AS1CAST(const_cast<float*>(wgt) + o * (CIN * KK * KK) + t),
            AS3CAST(sw + i), 0, 0);
#else
        sw[i] = wgt[o * (CIN * KK * KK) + t];
#endif
    }
#if USE_ASYNC_LDS
    // all of this wave's DMAs must land before the barrier releases readers
#if __has_builtin(__builtin_amdgcn_s_wait_asynccnt)
    __builtin_amdgcn_s_wait_asynccnt(0);
#else
    asm volatile("s_wait_asynccnt 0x0" ::: "memory");
#endif
#endif
    __syncthreads();

    // thread owns: 4 consecutive cols x 2 consecutive rows x 4 out-channels
    const int cg = tid & 15;              // 16 col groups  -> 64 cols
    const int rg = (tid >> 4) & 7;        // 8 row groups   -> 16 rows
    const int oh = tid >> 7;              // 2 o-halves     -> 8 chans
    const int cb = cg * 4;
    const int rb = rg * 2;
    const int ob = oh * 4;

    float acc[4][2][4];
#pragma unroll
    for (int o = 0; o < 4; ++o)
#pragma unroll
        for (int rr = 0; rr < 2; ++rr)
#pragma unroll
            for (int j = 0; j < 4; ++j)
                acc[o][rr][j] = -__builtin_inff();

#pragma unroll 1
    for (int c = 0; c < CIN; ++c) {
        const float* sxc = sx + c * (XROWS * XSL);
        const float* swc = sw + c * (KK * KK * COUT);
#pragma unroll 1
        for (int dy = 0; dy < KK; ++dy) {
            // 2 rows x 8 cols sliding window: 4x ds_load_b128
            const float4 a0 = *reinterpret_cast<const float4*>(sxc + (rb + dy)     * XSL + cb);
            const float4 a1 = *reinterpret_cast<const float4*>(sxc + (rb + dy)     * XSL + cb + 4);
            const float4 b0 = *reinterpret_cast<const float4*>(sxc + (rb + dy + 1) * XSL + cb);
            const float4 b1 = *reinterpret_cast<const float4*>(sxc + (rb + dy + 1) * XSL + cb + 4);
            const float xr[2][8] = {
                { a0.x, a0.y, a0.z, a0.w, a1.x, a1.y, a1.z, a1.w },
                { b0.x, b0.y, b0.z, b0.w, b1.x, b1.y, b1.z, b1.w }
            };
            const float* swd = swc + dy * (KK * COUT);
#pragma unroll
            for (int dx = 0; dx < KK; ++dx) {
                // 4 weights (one per owned o), broadcast ds_load_b128
                const float4 wv = *reinterpret_cast<const float4*>(swd + dx * COUT + ob);
                const float w4[4] = { wv.x, wv.y, wv.z, wv.w };
#pragma unroll
                for (int o = 0; o < 4; ++o)
#pragma unroll
                    for (int rr = 0; rr < 2; ++rr)
#pragma unroll
                        for (int j = 0; j < 4; ++j)
                            // scalar add+max stream: VOPD dual-issue eligible,
                            // fmaxf chains over dx fold to v_max3_num_f32
                            acc[o][rr][j] = fmaxf(acc[o][rr][j], xr[rr][j + dx] + w4[o]);
            }
        }
    }

    // ---- coalesced 16B stores ----
#pragma unroll
    for (int o = 0; o < 4; ++o)
#pragma unroll
        for (int rr = 0; rr < 2; ++rr) {
            float4 v = make_float4(acc[o][rr][0], acc[o][rr][1],
                                   acc[o][rr][2], acc[o][rr][3]);
            size_t off = (((size_t)n * COUT + (ob + o)) * HH + (row0 + rb + rr)) * WW
                         + (col0 + cb);
            *reinterpret_cast<float4*>(out + off) = v;
        }
}

extern "C" void kernel_launch(void* const* d_in, const int* in_sizes, int n_in,
                              void* d_out, int out_size, void* d_ws, size_t ws_size,
                              hipStream_t stream) {
    const float* x   = (const float*)d_in[0];
    const float* wgt = (const float*)d_in[1];
    float* out       = (float*)d_out;

    dim3 grid(WW / TILE_W, HH / TILE_H, 4);   // (8, 32, 4)
    dim3 block(256, 1, 1);                    // 8 wave32 per block
    dilation2d_maxplus<<<grid, block, 0, stream>>>(x, wgt, out);
}